// PositionFilter_28793460752638
// MI455X (gfx1250) — compile-verified
//
#include <hip/hip_runtime.h>
#include <hip/hip_bf16.h>
#include <stdint.h>

// ---------------------------------------------------------------------------
// Batched EKF step (predict + update), B ~ 1e6, state=6, obs=3.
// Memory-bound: ~532 B/tracklet -> ~23 us floor at 23.3 TB/s. WMMA is useless
// for 6x6 per-item algebra; the CDNA5 win is async global<->LDS staging for
// perfect coalescing of the 144 B/thread P/Q/P_upd streams. All streaming
// traffic is tagged non-temporal (read-once/write-once, no reuse).
// ---------------------------------------------------------------------------

#define TPB 128
#define WPB (TPB / 32)

// --- CDNA5 async global<->LDS ops (inline asm; gfx1250 VGLOBAL encodings) ---
__device__ __forceinline__ uint32_t lds_off(const void* p) {
  // generic pointer to LDS: low 32 bits == LDS byte offset
  return (uint32_t)(uintptr_t)p;
}
__device__ __forceinline__ void async_ld_b128(uint32_t lds, uint64_t gaddr) {
  asm volatile("global_load_async_to_lds_b128 %0, %1, off th:TH_LOAD_NT"
               :: "v"(lds), "v"(gaddr) : "memory");
}
__device__ __forceinline__ void async_ld_b32(uint32_t lds, uint64_t gaddr) {
  asm volatile("global_load_async_to_lds_b32 %0, %1, off th:TH_LOAD_NT"
               :: "v"(lds), "v"(gaddr) : "memory");
}
__device__ __forceinline__ void async_st_b128(uint64_t gaddr, uint32_t lds) {
  asm volatile("global_store_async_from_lds_b128 %0, %1, off th:TH_STORE_NT"
               :: "v"(gaddr), "v"(lds) : "memory");
}
__device__ __forceinline__ void async_st_b32(uint64_t gaddr, uint32_t lds) {
  asm volatile("global_store_async_from_lds_b32 %0, %1, off th:TH_STORE_NT"
               :: "v"(gaddr), "v"(lds) : "memory");
}
__device__ __forceinline__ void wait_async() {
  asm volatile("s_wait_asynccnt 0x0" ::: "memory");
}
__device__ __forceinline__ void wait_ds() {
  asm volatile("s_wait_dscnt 0x0" ::: "memory");
}

// --- per-tracklet EKF math, fully in registers --------------------------
// p[36]: in = P, out = P_pred (in-place). pu[36] = P_upd, xu[6] = x_upd.
__device__ __forceinline__ void ekf_step(float* __restrict__ p,
                                         const float* __restrict__ q,
                                         const float* __restrict__ r9,
                                         const float* __restrict__ xv,
                                         const float* __restrict__ zv,
                                         float dtv,
                                         float* __restrict__ xu,
                                         float* __restrict__ pu) {
  // P_pred = F P F^T + Q with F = I + dt*shift (shift: (0,3),(1,4),(2,5))
#pragma unroll
  for (int i = 0; i < 3; ++i) {
#pragma unroll
    for (int j = 0; j < 6; ++j) {
      float v = p[i * 6 + j] + dtv * p[(i + 3) * 6 + j];
      if (j < 3) v += dtv * (p[i * 6 + j + 3] + dtv * p[(i + 3) * 6 + j + 3]);
      p[i * 6 + j] = v + q[i * 6 + j];
    }
  }
#pragma unroll
  for (int i = 3; i < 6; ++i) {
#pragma unroll
    for (int j = 0; j < 6; ++j) {
      float v = p[i * 6 + j];
      if (j < 3) v += dtv * p[i * 6 + j + 3];
      p[i * 6 + j] = v + q[i * 6 + j];
    }
  }
  // S = P_pred[0:3,0:3] + R ; Sinv via adjugate
  float s00 = p[0] + r9[0],  s01 = p[1] + r9[1],  s02 = p[2] + r9[2];
  float s10 = p[6] + r9[3],  s11 = p[7] + r9[4],  s12 = p[8] + r9[5];
  float s20 = p[12] + r9[6], s21 = p[13] + r9[7], s22 = p[14] + r9[8];
  float c00 = s11 * s22 - s12 * s21;
  float c01 = s12 * s20 - s10 * s22;
  float c02 = s10 * s21 - s11 * s20;
  float det = s00 * c00 + s01 * c01 + s02 * c02;
  float id = 1.0f / det;
  float i00 = c00 * id;
  float i01 = (s02 * s21 - s01 * s22) * id;
  float i02 = (s01 * s12 - s02 * s11) * id;
  float i10 = c01 * id;
  float i11 = (s00 * s22 - s02 * s20) * id;
  float i12 = (s02 * s10 - s00 * s12) * id;
  float i20 = c02 * id;
  float i21 = (s01 * s20 - s00 * s21) * id;
  float i22 = (s00 * s11 - s01 * s10) * id;
  // K = P_pred[:, :3] @ Sinv  [6x3]
  float K[18];
#pragma unroll
  for (int i = 0; i < 6; ++i) {
    float a = p[i * 6 + 0], b = p[i * 6 + 1], c = p[i * 6 + 2];
    K[i * 3 + 0] = a * i00 + b * i10 + c * i20;
    K[i * 3 + 1] = a * i01 + b * i11 + c * i21;
    K[i * 3 + 2] = a * i02 + b * i12 + c * i22;
  }
  // x_pred, innovation, x_upd
  float xp[6];
#pragma unroll
  for (int i = 0; i < 3; ++i) xp[i] = xv[i] + dtv * xv[i + 3];
#pragma unroll
  for (int i = 3; i < 6; ++i) xp[i] = xv[i];
  float n0 = zv[0] - xp[0], n1 = zv[1] - xp[1], n2 = zv[2] - xp[2];
#pragma unroll
  for (int i = 0; i < 6; ++i)
    xu[i] = xp[i] + K[i * 3 + 0] * n0 + K[i * 3 + 1] * n1 + K[i * 3 + 2] * n2;
  // P_upd = P_pred - K @ P_pred[0:3, :]
#pragma unroll
  for (int i = 0; i < 6; ++i) {
#pragma unroll
    for (int j = 0; j < 6; ++j) {
      pu[i * 6 + j] = p[i * 6 + j] -
                      (K[i * 3 + 0] * p[j] + K[i * 3 + 1] * p[6 + j] +
                       K[i * 3 + 2] * p[12 + j]);
    }
  }
}

__global__ __launch_bounds__(TPB) void ekf_kernel(
    const float* __restrict__ gx, const float* __restrict__ gP,
    const float* __restrict__ gdt, const float* __restrict__ gQ,
    const float* __restrict__ gz, const float* __restrict__ gR,
    float* __restrict__ goX, float* __restrict__ goP, int Bn) {
  // per-wave private staging slices (no cross-wave sharing -> no barriers)
  __shared__ float sP[WPB * 32 * 36];
  __shared__ float sQ[WPB * 32 * 36];
  __shared__ float sR[WPB * 32 * 9];
  __shared__ float sX[WPB * 32 * 6];
  __shared__ float sZ[WPB * 32 * 3];

  const int lane = (int)(threadIdx.x & 31u);
  const int wid = (int)(threadIdx.x >> 5);

  float p[36], q[36], r9[9], xv[6], zv[3], xu[6], pu[36];

  if (Bn >= 32) {
    long long waveStart = (long long)blockIdx.x * TPB + (long long)wid * 32;
    if (waveStart >= (long long)Bn) return;  // uniform per wave
    long long eff = waveStart;
    if (eff + 32 > (long long)Bn) eff = (long long)Bn - 32;  // tail overlap
    const long long t = eff + lane;

    float* wP = &sP[wid * 32 * 36];
    float* wQ = &sQ[wid * 32 * 36];
    float* wR = &sR[wid * 32 * 9];
    float* wX = &sX[wid * 32 * 6];
    float* wZ = &sZ[wid * 32 * 3];
    const uint32_t lP = lds_off(wP);
    const uint32_t lQ = lds_off(wQ);
    const uint32_t lR = lds_off(wR);
    const uint32_t lX = lds_off(wX);
    const uint32_t lZ = lds_off(wZ);

    const uint64_t gPb = (uint64_t)(uintptr_t)(gP + eff * 36);
    const uint64_t gQb = (uint64_t)(uintptr_t)(gQ + eff * 36);
    const uint64_t gRb = (uint64_t)(uintptr_t)(gR + eff * 9);
    const uint64_t gXb = (uint64_t)(uintptr_t)(gx + eff * 6);
    const uint64_t gZb = (uint64_t)(uintptr_t)(gz + eff * 3);

    float dtv = __builtin_nontemporal_load(gdt + t);  // lane-coalesced

    // stage P & Q: 32 tracklets * 144 B = 4608 B each, linear b128 copy
#pragma unroll
    for (int k = 0; k < 9; ++k) {
      uint32_t off = (uint32_t)((k * 32 + lane) * 16);
      async_ld_b128(lP + off, gPb + off);
      async_ld_b128(lQ + off, gQb + off);
    }
    // stage R (1152 B), x (768 B), z (384 B): linear b32 copies
#pragma unroll
    for (int k = 0; k < 9; ++k) {
      uint32_t off = (uint32_t)((k * 32 + lane) * 4);
      async_ld_b32(lR + off, gRb + off);
    }
#pragma unroll
    for (int k = 0; k < 6; ++k) {
      uint32_t off = (uint32_t)((k * 32 + lane) * 4);
      async_ld_b32(lX + off, gXb + off);
    }
#pragma unroll
    for (int k = 0; k < 3; ++k) {
      uint32_t off = (uint32_t)((k * 32 + lane) * 4);
      async_ld_b32(lZ + off, gZb + off);
    }
    wait_async();  // this wave's copies landed in LDS

#pragma unroll
    for (int e = 0; e < 36; ++e) p[e] = wP[lane * 36 + e];
#pragma unroll
    for (int e = 0; e < 36; ++e) q[e] = wQ[lane * 36 + e];
#pragma unroll
    for (int e = 0; e < 9; ++e) r9[e] = wR[lane * 9 + e];
#pragma unroll
    for (int e = 0; e < 6; ++e) xv[e] = wX[lane * 6 + e];
#pragma unroll
    for (int e = 0; e < 3; ++e) zv[e] = wZ[lane * 3 + e];

    ekf_step(p, q, r9, xv, zv, dtv, xu, pu);

    // stage outputs into the now-free P and X slices, stream out coalesced
#pragma unroll
    for (int e = 0; e < 36; ++e) wP[lane * 36 + e] = pu[e];
#pragma unroll
    for (int e = 0; e < 6; ++e) wX[lane * 6 + e] = xu[e];
    wait_ds();  // DS writes committed before async-store engine reads LDS
    const uint64_t goPb = (uint64_t)(uintptr_t)(goP + eff * 36);
    const uint64_t goXb = (uint64_t)(uintptr_t)(goX + eff * 6);
#pragma unroll
    for (int k = 0; k < 9; ++k) {
      uint32_t off = (uint32_t)((k * 32 + lane) * 16);
      async_st_b128(goPb + off, lP + off);
    }
#pragma unroll
    for (int k = 0; k < 6; ++k) {
      uint32_t off = (uint32_t)((k * 32 + lane) * 4);
      async_st_b32(goXb + off, lX + off);
    }
    wait_async();  // don't let LDS be freed under an in-flight DMA
  } else {
    // tiny-batch fallback: direct global path
    long long t = (long long)blockIdx.x * TPB + (long long)threadIdx.x;
    if (t >= (long long)Bn) return;
    float dtv = gdt[t];
#pragma unroll
    for (int e = 0; e < 36; ++e) p[e] = gP[t * 36 + e];
#pragma unroll
    for (int e = 0; e < 36; ++e) q[e] = gQ[t * 36 + e];
#pragma unroll
    for (int e = 0; e < 9; ++e) r9[e] = gR[t * 9 + e];
#pragma unroll
    for (int e = 0; e < 6; ++e) xv[e] = gx[t * 6 + e];
#pragma unroll
    for (int e = 0; e < 3; ++e) zv[e] = gz[t * 3 + e];
    ekf_step(p, q, r9, xv, zv, dtv, xu, pu);
#pragma unroll
    for (int i = 0; i < 6; ++i) goX[t * 6 + i] = xu[i];
#pragma unroll
    for (int e = 0; e < 36; ++e) goP[t * 36 + e] = pu[e];
  }
}

extern "C" void kernel_launch(void* const* d_in, const int* in_sizes, int n_in,
                              void* d_out, int out_size, void* d_ws,
                              size_t ws_size, hipStream_t stream) {
  (void)n_in; (void)out_size; (void)d_ws; (void)ws_size;
  const float* x = (const float*)d_in[0];   // [B,6,1]
  const float* P = (const float*)d_in[1];   // [B,6,6]
  const float* dt = (const float*)d_in[2];  // [B]
  const float* Q = (const float*)d_in[3];   // [B,6,6]
  const float* z = (const float*)d_in[4];   // [B,3,1]
  const float* R = (const float*)d_in[5];   // [B,3,3]
  const int Bn = in_sizes[0] / 6;
  float* outX = (float*)d_out;              // [B,6,1]
  float* outP = outX + (size_t)Bn * 6;      // [B,6,6]
  const int blocks = (Bn + TPB - 1) / TPB;
  ekf_kernel<<<blocks, TPB, 0, stream>>>(x, P, dt, Q, z, R, outX, outP, Bn);
}